// LocalAffineAlignment_24824910970941
// MI455X (gfx1250) — compile-verified
//
#include <hip/hip_runtime.h>
#include <cstdint>

// ---------------------------------------------------------------------------
// LocalAffineAlignment for MI455X (gfx1250, wave32).
// Bandwidth-bound (~236 MB traffic -> ~10us @ 23.3 TB/s). Uses CDNA5 async
// global->LDS data movement (global_load_async_to_lds_b128 / s_wait_asynccnt)
// to stage each 16-patch tile once, then computes two-pass masked regression
// stats per patch with wave32 shfl reductions, and writes float4-coalesced
// outputs.
// ---------------------------------------------------------------------------

namespace {

constexpr int PS  = 8;            // patch size
constexpr int CH  = 3;            // channels
constexpr int Bn  = 16;           // batch
constexpr int Hn  = 480;
constexpr int Wn  = 640;
constexpr int NH  = Hn / PS;      // 60 patch rows
constexpr int TPB = 16;           // patches per block (along W)
constexpr int TILE_W = TPB * PS;  // 128 pixels
constexpr int TILES_X = Wn / TILE_W; // 5
constexpr float EPS = 1e-6f;
constexpr float ALPHA_MIN = 0.01f;
constexpr float MIN_VALID = 19.0f; // int(0.1 * 3 * 64) = 19

constexpr int T_FLOATS = CH * PS * TILE_W;  // 3072 floats per image tile
constexpr int M_FLOATS = PS * TILE_W;       // 1024 floats mask tile

// Issue one 16-byte async DMA: global -> LDS. lds_byte is the LDS byte
// address (low 32 bits of the generic pointer == LDS offset, ISA 10.2).
__device__ __forceinline__ void async_cp_b128(uint32_t lds_byte, const float* g) {
  asm volatile("global_load_async_to_lds_b128 %0, %1, off"
               :: "v"(lds_byte), "v"((unsigned long long)(uintptr_t)g)
               : "memory");
}

__device__ __forceinline__ void wait_async_all() {
  asm volatile("s_wait_asynccnt 0" ::: "memory");
}

__global__ __launch_bounds__(256)
void laa_kernel(const float* __restrict__ tgt,
                const float* __restrict__ src,
                const float* __restrict__ msk,
                float* __restrict__ out_aligned,
                float* __restrict__ out_alpha,
                float* __restrict__ out_beta) {
  __shared__ __align__(16) float smem[2 * T_FLOATS + M_FLOATS + 2 * TPB];
  float* s_t = smem;                       // [CH][PS][TILE_W]
  float* s_s = smem + T_FLOATS;            // [CH][PS][TILE_W]
  float* s_m = smem + 2 * T_FLOATS;        // [PS][TILE_W]
  float* s_a = smem + 2 * T_FLOATS + M_FLOATS;   // [TPB] alpha
  float* s_b = s_a + TPB;                        // [TPB] beta

  const int tid = threadIdx.x;
  int bid = blockIdx.x;
  const int tx = bid % TILES_X; bid /= TILES_X;
  const int ph = bid % NH;      bid /= NH;
  const int b  = bid;

  const int y0 = ph * PS;
  const int x0 = tx * TILE_W;

  const uint32_t lds_base = (uint32_t)(uintptr_t)(void*)smem;

  // ---------------- Phase 1: async stage tile into LDS ----------------
#pragma unroll
  for (int i = 0; i < 3; ++i) {
    const int f  = i * 256 + tid;          // float4 index, [0, 768)
    const int fi = f * 4;                  // float index,  [0, 3072)
    const int cr = fi >> 7;                // c*8 + r
    const int c  = cr >> 3;
    const int r  = cr & 7;
    const int x4 = fi & (TILE_W - 1);
    const int gofs = ((b * CH + c) * Hn + (y0 + r)) * Wn + x0 + x4;
    async_cp_b128(lds_base + (uint32_t)(f * 16), tgt + gofs);
    async_cp_b128(lds_base + (uint32_t)(T_FLOATS * 4 + f * 16), src + gofs);
  }
  {
    const int fi = tid * 4;                // [0, 1024)
    const int r  = fi >> 7;
    const int x4 = fi & (TILE_W - 1);
    const int gofs = (b * Hn + (y0 + r)) * Wn + x0 + x4;
    async_cp_b128(lds_base + (uint32_t)(2 * T_FLOATS * 4 + tid * 16), msk + gofs);
  }
  wait_async_all();       // this wave's DMAs complete
  __syncthreads();        // all waves' DMAs visible

  // ---------------- Phase 2: per-patch stats (16 lanes / patch) --------
  const int p   = tid >> 4;   // patch 0..15
  const int sub = tid & 15;   // lane within patch group

  float sv = 0.f, stv = 0.f, ssv = 0.f;
#pragma unroll
  for (int j = 0; j < 12; ++j) {
    const int e   = sub + j * 16;          // element 0..191 (c*64 + k)
    const int c   = e >> 6;
    const int k   = e & 63;
    const int r   = k >> 3;
    const int col = (k & 7) + p * PS;
    const float v = s_m[r * TILE_W + col];
    const float t = s_t[(c * PS + r) * TILE_W + col];
    const float s = s_s[(c * PS + r) * TILE_W + col];
    sv  += v;
    stv += v * t;
    ssv += v * s;
  }
#pragma unroll
  for (int m = 8; m; m >>= 1) {
    sv  += __shfl_xor(sv,  m, 32);
    stv += __shfl_xor(stv, m, 32);
    ssv += __shfl_xor(ssv, m, 32);
  }
  const float denom = sv + EPS;
  const float tmean = stv / denom;
  const float smean = ssv / denom;

  float cov = 0.f, var = 0.f;
#pragma unroll
  for (int j = 0; j < 12; ++j) {
    const int e   = sub + j * 16;
    const int c   = e >> 6;
    const int k   = e & 63;
    const int r   = k >> 3;
    const int col = (k & 7) + p * PS;
    const float v = s_m[r * TILE_W + col];
    const float t = s_t[(c * PS + r) * TILE_W + col];
    const float s = s_s[(c * PS + r) * TILE_W + col];
    const float tc = (t - tmean) * v;
    const float sc = (s - smean) * v;
    cov += tc * sc;
    var += sc * sc;
  }
#pragma unroll
  for (int m = 8; m; m >>= 1) {
    cov += __shfl_xor(cov, m, 32);
    var += __shfl_xor(var, m, 32);
  }

  if (sub == 0) {
    const float covd = cov / denom;
    const float vard = var / denom;
    float alpha = fmaxf(covd / (vard + EPS), ALPHA_MIN);
    float beta  = tmean - alpha * smean;           // uses clamped alpha (pre-blend)
    const float pv = (sv >= MIN_VALID) ? 1.f : 0.f;
    alpha = alpha * pv + (1.f - pv);
    beta  = beta * pv;
    s_a[p] = alpha;
    s_b[p] = beta;
  }
  __syncthreads();

  // ---------------- Phase 3: apply + write outputs (float4) ------------
#pragma unroll
  for (int i = 0; i < 3; ++i) {
    const int f  = i * 256 + tid;
    const int fi = f * 4;
    const int cr = fi >> 7;
    const int c  = cr >> 3;
    const int r  = cr & 7;
    const int x4 = fi & (TILE_W - 1);
    const int pp = x4 >> 3;                // 4-aligned chunk never crosses a patch
    const float a  = s_a[pp];
    const float bt = s_b[pp];
    const float4 sv4 = *reinterpret_cast<const float4*>(&s_s[cr * TILE_W + x4]);
    float4 o;
    o.x = fmaf(a, sv4.x, bt);
    o.y = fmaf(a, sv4.y, bt);
    o.z = fmaf(a, sv4.z, bt);
    o.w = fmaf(a, sv4.w, bt);
    const int gofs = ((b * CH + c) * Hn + (y0 + r)) * Wn + x0 + x4;
    *reinterpret_cast<float4*>(out_aligned + gofs) = o;
  }
  {
    const int fi = tid * 4;
    const int r  = fi >> 7;
    const int x4 = fi & (TILE_W - 1);
    const int pp = x4 >> 3;
    const float a  = s_a[pp];
    const float bt = s_b[pp];
    const int gofs = (b * Hn + (y0 + r)) * Wn + x0 + x4;
    *reinterpret_cast<float4*>(out_alpha + gofs) = make_float4(a, a, a, a);
    *reinterpret_cast<float4*>(out_beta + gofs)  = make_float4(bt, bt, bt, bt);
  }
}

} // namespace

extern "C" void kernel_launch(void* const* d_in, const int* in_sizes, int n_in,
                              void* d_out, int out_size, void* d_ws, size_t ws_size,
                              hipStream_t stream) {
  const float* tgt = (const float*)d_in[0];   // target_img      (B,C,H,W)
  const float* src = (const float*)d_in[1];   // warped_source   (B,C,H,W)
  const float* msk = (const float*)d_in[2];   // valid_mask      (B,1,H,W)

  float* out         = (float*)d_out;
  float* out_aligned = out;                                   // (B,C,H,W)
  float* out_alpha   = out + (size_t)Bn * CH * Hn * Wn;       // (B,1,H,W)
  float* out_beta    = out_alpha + (size_t)Bn * Hn * Wn;      // (B,1,H,W)

  const dim3 grid(Bn * NH * TILES_X);  // 16*60*5 = 4800 blocks
  const dim3 block(256);               // 8 wave32 waves
  laa_kernel<<<grid, block, 0, stream>>>(tgt, src, msk,
                                         out_aligned, out_alpha, out_beta);
}